// IF_80702435492066
// MI455X (gfx1250) — compile-verified
//
#include <hip/hip_runtime.h>

typedef float v4f __attribute__((ext_vector_type(4)));

#define TSTEPS 3
#define BLOCK 256
#define VEC 4
#define TILE_ELEMS (BLOCK * VEC)   // 1024 floats per timestep per tile
#define MAX_BLOCKS 4096

// Double-buffered async HBM->LDS pipeline. Each thread handles one float4 per
// timestep per tile; async loads tracked with ASYNCcnt, waited per-wave.
__global__ __launch_bounds__(BLOCK) void if_spike_vec_kernel(
    const float* __restrict__ x,
    const float* __restrict__ alpha,
    float* __restrict__ out,
    long long N, long long ntiles)
{
    __shared__ v4f smem[2][TSTEPS][BLOCK];   // 24 KB

    const float th = alpha[0];
    const float half_th = 0.5f * th;
    const unsigned tid = threadIdx.x;

    // Low 32 bits of a generic pointer to LDS are the LDS byte offset.
    const unsigned lds_base =
        (unsigned)(unsigned long long)(&smem[0][0][0]) + tid * (unsigned)sizeof(v4f);
    const unsigned buf_stride = TSTEPS * BLOCK * (unsigned)sizeof(v4f); // 12288 B
    const unsigned t_stride   = BLOCK * (unsigned)sizeof(v4f);          // 4096 B

    long long tile = blockIdx.x;
    const long long tile_step = gridDim.x;

    // Prologue: prefetch first tile into buffer 0.
    if (tile < ntiles) {
        const long long ebase = tile * TILE_ELEMS + (long long)tid * VEC;
#pragma unroll
        for (int t = 0; t < TSTEPS; ++t) {
            const float* g = x + (long long)t * N + ebase;
            unsigned l = lds_base + (unsigned)t * t_stride;
            asm volatile("global_load_async_to_lds_b128 %0, %1, off"
                         :: "v"(l), "v"(g) : "memory");
        }
    }

    unsigned parity = 0;
    for (; tile < ntiles; tile += tile_step, parity ^= 1) {
        const long long next = tile + tile_step;
        if (next < ntiles) {
            // Prefetch next tile into the other buffer, then wait so that only
            // the 3 just-issued loads may remain outstanding (in-order done).
            const long long ebase = next * TILE_ELEMS + (long long)tid * VEC;
            const unsigned lbase = lds_base + (parity ^ 1u) * buf_stride;
#pragma unroll
            for (int t = 0; t < TSTEPS; ++t) {
                const float* g = x + (long long)t * N + ebase;
                unsigned l = lbase + (unsigned)t * t_stride;
                asm volatile("global_load_async_to_lds_b128 %0, %1, off"
                             :: "v"(l), "v"(g) : "memory");
            }
            asm volatile("s_wait_asynccnt 0x3" ::: "memory");
        } else {
            asm volatile("s_wait_asynccnt 0x0" ::: "memory");
        }

        const v4f a = smem[parity][0][tid];
        const v4f b = smem[parity][1][tid];
        const v4f c = smem[parity][2][tid];

        v4f s0, s1, s2;
#pragma unroll
        for (int k = 0; k < VEC; ++k) {
            float m  = half_th + a[k] + b[k] + c[k];
            float p0 = (m > th) ? th : 0.0f; m -= p0;
            float p1 = (m > th) ? th : 0.0f; m -= p1;
            float p2 = (m > th) ? th : 0.0f;
            s0[k] = p0; s1[k] = p1; s2[k] = p2;
        }

        const long long ebase = tile * TILE_ELEMS + (long long)tid * VEC;
        __builtin_nontemporal_store(s0, (v4f*)(out + ebase));
        __builtin_nontemporal_store(s1, (v4f*)(out + N + ebase));
        __builtin_nontemporal_store(s2, (v4f*)(out + 2 * N + ebase));
    }
}

// Scalar tail for N % TILE_ELEMS (empty for the reference shape).
__global__ void if_spike_tail_kernel(const float* __restrict__ x,
                                     const float* __restrict__ alpha,
                                     float* __restrict__ out,
                                     long long N, long long start)
{
    long long i = start + (long long)blockIdx.x * blockDim.x + threadIdx.x;
    if (i >= N) return;
    const float th = alpha[0];
    float m  = 0.5f * th + x[i] + x[N + i] + x[2 * N + i];
    float p0 = (m > th) ? th : 0.0f; m -= p0;
    float p1 = (m > th) ? th : 0.0f; m -= p1;
    float p2 = (m > th) ? th : 0.0f;
    out[i] = p0; out[N + i] = p1; out[2 * N + i] = p2;
}

extern "C" void kernel_launch(void* const* d_in, const int* in_sizes, int n_in,
                              void* d_out, int out_size, void* d_ws, size_t ws_size,
                              hipStream_t stream) {
    const float* x     = (const float*)d_in[0];
    const float* alpha = (const float*)d_in[1];
    float* out         = (float*)d_out;

    const long long total = (long long)in_sizes[0];  // T * N
    const long long N     = total / TSTEPS;          // elements per timestep
    const long long ntiles = N / TILE_ELEMS;

    if (ntiles > 0) {
        int blocks = (int)(ntiles < (long long)MAX_BLOCKS ? ntiles : (long long)MAX_BLOCKS);
        if_spike_vec_kernel<<<blocks, BLOCK, 0, stream>>>(x, alpha, out, N, ntiles);
    }

    const long long start = ntiles * (long long)TILE_ELEMS;
    const long long rem   = N - start;
    if (rem > 0) {
        const int tb = 256;
        const int gb = (int)((rem + tb - 1) / tb);
        if_spike_tail_kernel<<<gb, tb, 0, stream>>>(x, alpha, out, N, start);
    }
}